// InceptionModule_72052371358072
// MI455X (gfx1250) — compile-verified
//
#include <hip/hip_runtime.h>
#include <hip/hip_bf16.h>

// ---------------------------------------------------------------------------
// InceptionModule for MI455X (gfx1250): f32 WMMA (16x16x4) for the conv GEMMs
// and the 640x640 bottleneck GEMM; async global->LDS staging; small fused
// kernels for GAP/MHSA/SE/BN.
// ---------------------------------------------------------------------------

typedef __attribute__((ext_vector_type(2))) float v2f;
typedef __attribute__((ext_vector_type(8))) float v8f;

#define B_SZ   32
#define NI     128
#define NF     128
#define LLEN   4096
#define NFT    5
#define NC     640           // NFT*NF
#define EPS    1e-5f

// workspace layout (in floats)
#define OFF_F     0ul                          // F / z region: 5*32*128*4096 = 83886080
#define OFF_GAPS  83886080ul                   // 32*5*128 = 20480
#define OFF_ATT   83906560ul                   // 32*25    = 800
#define OFF_Y     83907360ul                   // 32*640   = 20480
#define OFF_STATS 83927840ul                   // 2*640    = 1280

static __device__ __forceinline__ v8f wmma_f32(v2f a, v2f b, v8f c) {
  // D = A(16x4, f32) * B(4x16, f32) + C(16x16, f32)
  return __builtin_amdgcn_wmma_f32_16x16x4_f32(
      false, a, false, b, (short)0, c, false, false);
}

// ----- CDNA5 async global->LDS staging (ASYNCcnt path), with safe fallback --
#if defined(__has_builtin)
#if __has_builtin(__builtin_amdgcn_global_load_async_to_lds_b32)
#define HAVE_ASYNC_LDS 1
#endif
#endif

typedef __attribute__((address_space(1))) int GASInt;   // global (AS1)
typedef __attribute__((address_space(3))) int LASInt;   // LDS    (AS3)

static __device__ __forceinline__ void stage_b32(const float* gsrc, float* ldst) {
#ifdef HAVE_ASYNC_LDS
  __builtin_amdgcn_global_load_async_to_lds_b32(
      (GASInt*)(uintptr_t)gsrc, (LASInt*)(uintptr_t)ldst, 0, 0);
#else
  *ldst = *gsrc;
#endif
}

static __device__ __forceinline__ void stage_wait() {
#ifdef HAVE_ASYNC_LDS
  asm volatile("s_wait_asynccnt 0" ::: "memory");
#endif
}

// ---------------------------------------------------------------------------
// Kernel 1: 5 conv branches -> F[5][B][128][4096]
// block: 128 threads (4 waves), wave tile = 32M x 64N, block = [128 x 64]
// LDS: x tile transposed (pairs along channel), weights K-pair interleaved.
// grid: (L/64, B, 5)
// ---------------------------------------------------------------------------
__global__ __launch_bounds__(128) void conv5_kernel(
    const float* __restrict__ x,
    const float* __restrict__ w0, const float* __restrict__ w1,
    const float* __restrict__ w2, const float* __restrict__ w3,
    const float* __restrict__ wmp,
    float* __restrict__ Fout)
{
  __shared__ float sXt[82][130];   // [col(=64+18 halo)][channel], pairs contiguous
  __shared__ float sWA[16][264];   // [k-pair][2*o + (k&1)] : A frags as b64

  const int tid  = threadIdx.x;
  const int lane = tid & 31;
  const int wv   = tid >> 5;      // 0..3 -> M offset 32*wv
  const int r    = lane & 15;
  const int half = lane >> 4;

  const int l0 = blockIdx.x * 64;
  const int b  = blockIdx.y;
  const int br = blockIdx.z;

  const float* W = (br == 0) ? w0 : (br == 1) ? w1 : (br == 2) ? w2
                   : (br == 3) ? w3 : wmp;
  const int kw  = (br == 0) ? 19 : (br == 1) ? 9 : (br == 2) ? 5 : 1;
  const int pad = kw >> 1;

  // stage x tile (transposed) with +/-9 halo; async for in-range lanes
  for (int idx = tid; idx < 128 * 82; idx += 128) {
    int i = idx / 82, j = idx % 82;          // i = channel, j = halo column
    int g = l0 - 9 + j;
    if (g >= 0 && g < LLEN)
      stage_b32(&x[((size_t)b * NI + i) * LLEN + g], &sXt[j][i]);
    else
      sXt[j][i] = 0.0f;
  }

  const int m0 = wv * 32;

  v8f acc[2][4];
#pragma unroll
  for (int mt = 0; mt < 2; ++mt)
#pragma unroll
    for (int nt = 0; nt < 4; ++nt)
      acc[mt][nt] = (v8f){0,0,0,0,0,0,0,0};

  auto pool_pair = [&](int ci, int j, int g) -> v2f {
    v2f m = *reinterpret_cast<const v2f*>(&sXt[j][ci]);
    if (g > 0) {
      v2f t = *reinterpret_cast<const v2f*>(&sXt[j - 1][ci]);
      m.x = fmaxf(m.x, t.x); m.y = fmaxf(m.y, t.y);
    }
    if (g < LLEN - 1) {
      v2f t = *reinterpret_cast<const v2f*>(&sXt[j + 1][ci]);
      m.x = fmaxf(m.x, t.x); m.y = fmaxf(m.y, t.y);
    }
    return m;
  };

  for (int dk = 0; dk < kw; ++dk) {
    for (int kc = 0; kc < 128; kc += 32) {
      stage_wait();
      __syncthreads();
      // stage K-pair-interleaved weight chunk: sWA[kk>>1][2*o+(kk&1)] = W[o][kc+kk][dk]
      for (int idx = tid; idx < 32 * 128; idx += 128) {
        int o = idx >> 5, kk = idx & 31;
        stage_b32(&W[((size_t)o * 128 + (kc + kk)) * kw + dk],
                  &sWA[kk >> 1][2 * o + (kk & 1)]);
      }
      stage_wait();
      __syncthreads();
#pragma unroll
      for (int k4 = 0; k4 < 32; k4 += 4) {
        const int p  = (k4 >> 1) + half;     // K-pair index for this lane half
        const int ci = kc + k4 + 2 * half;   // channel pair base (even)
        v2f a0 = *reinterpret_cast<const v2f*>(&sWA[p][2 * (m0 + r)]);
        v2f a1 = *reinterpret_cast<const v2f*>(&sWA[p][2 * (m0 + 16 + r)]);
        v2f bf[4];
#pragma unroll
        for (int nt = 0; nt < 4; ++nt) {
          const int col = nt * 16 + r;
          if (br < 4) {
            const int j = col + 9 + dk - pad;
            bf[nt] = *reinterpret_cast<const v2f*>(&sXt[j][ci]);
          } else {
            bf[nt] = pool_pair(ci, col + 9, l0 + col);
          }
        }
#pragma unroll
        for (int nt = 0; nt < 4; ++nt) {
          acc[0][nt] = wmma_f32(a0, bf[nt], acc[0][nt]);
          acc[1][nt] = wmma_f32(a1, bf[nt], acc[1][nt]);
        }
      }
    }
  }

  // store: C/D layout -> VGPR v holds row (v + 8*half), col = lane%16
#pragma unroll
  for (int mt = 0; mt < 2; ++mt) {
#pragma unroll
    for (int v = 0; v < 8; ++v) {
      const int row = m0 + mt * 16 + v + 8 * half;
      const size_t base = ((size_t)(br * B_SZ + b) * NF + row) * LLEN + l0 + r;
#pragma unroll
      for (int nt = 0; nt < 4; ++nt)
        Fout[base + nt * 16] = acc[mt][nt][v];
    }
  }
}

// ---------------------------------------------------------------------------
// Kernel 2: GAP  gaps[b][br][c] = mean_l F[br][b][c][l]
// grid: (128, 32, 5), 256 threads
// ---------------------------------------------------------------------------
__global__ __launch_bounds__(256) void gap_kernel(
    const float* __restrict__ F, float* __restrict__ gaps)
{
  __shared__ float red[256];
  const int c = blockIdx.x, b = blockIdx.y, br = blockIdx.z;
  const size_t base = ((size_t)(br * B_SZ + b) * NF + c) * LLEN;
  float s = 0.0f;
  for (int l = threadIdx.x; l < LLEN; l += 256) s += F[base + l];
  red[threadIdx.x] = s;
  __syncthreads();
  for (int off = 128; off > 0; off >>= 1) {
    if (threadIdx.x < off) red[threadIdx.x] += red[threadIdx.x + off];
    __syncthreads();
  }
  if (threadIdx.x == 0)
    gaps[((size_t)b * NFT + br) * NF + c] = red[0] * (1.0f / LLEN);
}

// ---------------------------------------------------------------------------
// Kernel 3: per-batch MHSA over 5 branch tokens + SE gates
// grid: 32 blocks, 128 threads
// ---------------------------------------------------------------------------
__global__ __launch_bounds__(128) void attse_kernel(
    const float* __restrict__ gaps,
    const float* __restrict__ qkv_w, const float* __restrict__ qkv_b,
    const float* __restrict__ se_w1, const float* __restrict__ se_w2,
    float* __restrict__ att_out, float* __restrict__ y_out)
{
  __shared__ float g_s[NFT][NF];
  __shared__ float qkv_s[NFT][3 * NF];
  __shared__ float e_s[4][NFT][NFT];
  __shared__ float att_s[NFT][NFT];
  __shared__ float yp_s[NC];
  __shared__ float h1_s[40];

  const int b = blockIdx.x;
  const int tid = threadIdx.x;

  for (int t = 0; t < NFT; ++t)
    g_s[t][tid] = gaps[((size_t)b * NFT + t) * NF + tid];
  __syncthreads();

  // qkv = gaps @ qkv_w^T + qkv_b     (qkv_w: [384,128])
  for (int o = tid; o < 3 * NF; o += 128) {
    const float* wr = qkv_w + (size_t)o * NF;
    float bias = qkv_b[o];
    for (int t = 0; t < NFT; ++t) {
      float s = bias;
      for (int d = 0; d < NF; ++d) s += g_s[t][d] * wr[d];
      qkv_s[t][o] = s;
    }
  }
  __syncthreads();

  // per-token layout after reshape(5, H=4, dh=32, 3): idx = h*96 + d*3 + sel
  if (tid < 100) {
    int h = tid / 25, qi = (tid % 25) / 5, kj = tid % 5;
    float e = 0.0f;
    for (int d = 0; d < 32; ++d)
      e += qkv_s[qi][h * 96 + d * 3 + 0] * qkv_s[kj][h * 96 + d * 3 + 1];
    e_s[h][qi][kj] = e;
  }
  __syncthreads();

  if (tid < 20) {   // softmax over kj, scaled by 1/sqrt(128)
    int h = tid / 5, qi = tid % 5;
    const float inv = 1.0f / sqrtf((float)NF);
    float m = -3.4e38f;
    for (int kj = 0; kj < NFT; ++kj) m = fmaxf(m, e_s[h][qi][kj] * inv);
    float sum = 0.0f, p[NFT];
    for (int kj = 0; kj < NFT; ++kj) {
      p[kj] = __expf(e_s[h][qi][kj] * inv - m);
      sum += p[kj];
    }
    for (int kj = 0; kj < NFT; ++kj) e_s[h][qi][kj] = p[kj] / sum;
  }
  __syncthreads();

  if (tid < 25) {   // head average
    int qi = tid / 5, kj = tid % 5;
    float a = 0.25f * (e_s[0][qi][kj] + e_s[1][qi][kj] +
                       e_s[2][qi][kj] + e_s[3][qi][kj]);
    att_s[qi][kj] = a;
    att_out[(size_t)b * 25 + tid] = a;
  }
  __syncthreads();

  // SE input: mean_l(xc) = att . gaps (mixing commutes with the L-mean)
  for (int i = 0; i < NFT; ++i) {
    float s = 0.0f;
    for (int j = 0; j < NFT; ++j) s += att_s[i][j] * g_s[j][tid];
    yp_s[i * NF + tid] = s;
  }
  __syncthreads();

  if (tid < 40) {   // se_w1: [40, 640]
    float s = 0.0f;
    const float* wr = se_w1 + (size_t)tid * NC;
    for (int m = 0; m < NC; ++m) s += yp_s[m] * wr[m];
    h1_s[tid] = fmaxf(s, 0.0f);
  }
  __syncthreads();

  for (int o = tid; o < NC; o += 128) {  // se_w2: [640, 40]
    float s = 0.0f;
    const float* wr = se_w2 + (size_t)o * 40;
    for (int m = 0; m < 40; ++m) s += h1_s[m] * wr[m];
    y_out[(size_t)b * NC + o] = 1.0f / (1.0f + __expf(-s));
  }
}

// ---------------------------------------------------------------------------
// Kernel 4: xc[b][i*128+c][l] = y * sum_j att[i][j] * F[j][b][c][l]  -> d_out
// ---------------------------------------------------------------------------
__global__ __launch_bounds__(256) void xcbuild_kernel(
    const float* __restrict__ F, const float* __restrict__ att,
    const float* __restrict__ y, float* __restrict__ xc)
{
  __shared__ float s_att[25];
  __shared__ float s_y[NFT];
  const int tid = threadIdx.x;
  const size_t gid = (size_t)blockIdx.x * 256 + tid;
  const int b  = (int)(gid >> 19);           // 128*4096 = 2^19 per batch
  const int rem = (int)(gid & 524287);
  const int cc = rem >> 12;
  const int l  = rem & 4095;
  if (tid < 25) s_att[tid] = att[(size_t)b * 25 + tid];
  if (tid < NFT) s_y[tid]  = y[(size_t)b * NC + tid * NF + cc];
  __syncthreads();

  float f[NFT];
#pragma unroll
  for (int j = 0; j < NFT; ++j)
    f[j] = F[((size_t)(j * B_SZ + b) * NF + cc) * LLEN + l];
#pragma unroll
  for (int i = 0; i < NFT; ++i) {
    float a = 0.0f;
#pragma unroll
    for (int j = 0; j < NFT; ++j) a += s_att[i * 5 + j] * f[j];
    xc[((size_t)b * NC + i * NF + cc) * LLEN + l] = s_y[i] * a;
  }
}

// ---------------------------------------------------------------------------
// Kernel 5: bottleneck  z[b][o][l] = sum_c Wb[o][c] * xc[b][c][l]
// block: 128 threads (4 waves), wave tile 32M x 64N; K=640; grid (64, 32, 5)
// ---------------------------------------------------------------------------
__global__ __launch_bounds__(128) void bott_kernel(
    const float* __restrict__ xc, const float* __restrict__ Wb,
    float* __restrict__ z)
{
  __shared__ float sBt[64][130];   // [col][channel-in-chunk], pairs contiguous
  __shared__ float sWA[16][264];   // [k-pair][2*o + (k&1)]

  const int tid  = threadIdx.x;
  const int lane = tid & 31;
  const int wv   = tid >> 5;
  const int r    = lane & 15;
  const int half = lane >> 4;

  const int l0 = blockIdx.x * 64;
  const int b  = blockIdx.y;
  const int o0 = blockIdx.z * 128;
  const int m0 = wv * 32;

  v8f acc[2][4];
#pragma unroll
  for (int mt = 0; mt < 2; ++mt)
#pragma unroll
    for (int nt = 0; nt < 4; ++nt)
      acc[mt][nt] = (v8f){0,0,0,0,0,0,0,0};

  for (int kb = 0; kb < NC; kb += 128) {
    stage_wait();
    __syncthreads();
    for (int idx = tid; idx < 128 * 64; idx += 128) {
      int c = idx >> 6, n = idx & 63;
      stage_b32(&xc[((size_t)b * NC + kb + c) * LLEN + l0 + n], &sBt[n][c]);
    }
    for (int kc = 0; kc < 128; kc += 32) {
      stage_wait();
      __syncthreads();
      for (int idx = tid; idx < 32 * 128; idx += 128) {
        int o = idx >> 5, kk = idx & 31;
        stage_b32(&Wb[(size_t)(o0 + o) * NC + kb + kc + kk],
                  &sWA[kk >> 1][2 * o + (kk & 1)]);
      }
      stage_wait();
      __syncthreads();
#pragma unroll
      for (int k4 = 0; k4 < 32; k4 += 4) {
        const int p  = (k4 >> 1) + half;
        const int ci = kc + k4 + 2 * half;
        v2f a0 = *reinterpret_cast<const v2f*>(&sWA[p][2 * (m0 + r)]);
        v2f a1 = *reinterpret_cast<const v2f*>(&sWA[p][2 * (m0 + 16 + r)]);
        v2f bf[4];
#pragma unroll
        for (int nt = 0; nt < 4; ++nt)
          bf[nt] = *reinterpret_cast<const v2f*>(&sBt[nt * 16 + r][ci]);
#pragma unroll
        for (int nt = 0; nt < 4; ++nt) {
          acc[0][nt] = wmma_f32(a0, bf[nt], acc[0][nt]);
          acc[1][nt] = wmma_f32(a1, bf[nt], acc[1][nt]);
        }
      }
    }
  }

#pragma unroll
  for (int mt = 0; mt < 2; ++mt) {
#pragma unroll
    for (int v = 0; v < 8; ++v) {
      const int row = o0 + m0 + mt * 16 + v + 8 * half;
      const size_t base = ((size_t)b * NC + row) * LLEN + l0 + r;
#pragma unroll
      for (int nt = 0; nt < 4; ++nt)
        z[base + nt * 16] = acc[mt][nt][v];
    }
  }
}

// ---------------------------------------------------------------------------
// Kernel 5b: per-channel batch stats (sum, sumsq) — one block per channel
// ---------------------------------------------------------------------------
__global__ __launch_bounds__(256) void bnstats_kernel(
    const float* __restrict__ z, float* __restrict__ stats)
{
  __shared__ float rs[256];
  __shared__ float rs2[256];
  const int o = blockIdx.x;
  float s = 0.0f, s2 = 0.0f;
  for (int idx = threadIdx.x; idx < B_SZ * LLEN; idx += 256) {
    int b = idx >> 12, l = idx & 4095;
    float v = z[((size_t)b * NC + o) * LLEN + l];
    s += v; s2 += v * v;
  }
  rs[threadIdx.x] = s; rs2[threadIdx.x] = s2;
  __syncthreads();
  for (int off = 128; off > 0; off >>= 1) {
    if (threadIdx.x < off) {
      rs[threadIdx.x]  += rs[threadIdx.x + off];
      rs2[threadIdx.x] += rs2[threadIdx.x + off];
    }
    __syncthreads();
  }
  if (threadIdx.x == 0) { stats[o] = rs[0]; stats[NC + o] = rs2[0]; }
}

// ---------------------------------------------------------------------------
// Kernel 6: BN (batch stats) + ReLU -> d_out
// ---------------------------------------------------------------------------
__global__ __launch_bounds__(256) void bnfinal_kernel(
    const float* __restrict__ z, const float* __restrict__ stats,
    const float* __restrict__ gamma, const float* __restrict__ beta,
    float* __restrict__ out)
{
  const size_t gid = (size_t)blockIdx.x * 256 + threadIdx.x;
  const int o = (int)((gid >> 12) % NC);
  const float inv_n = 1.0f / (float)(B_SZ * LLEN);
  float mean = stats[o] * inv_n;
  float var  = stats[NC + o] * inv_n - mean * mean;
  float sc   = rsqrtf(var + EPS) * gamma[o];
  float v = (z[gid] - mean) * sc + beta[o];
  out[gid] = fmaxf(v, 0.0f);
}

// ---------------------------------------------------------------------------
extern "C" void kernel_launch(void* const* d_in, const int* in_sizes, int n_in,
                              void* d_out, int out_size, void* d_ws, size_t ws_size,
                              hipStream_t stream) {
  const float* x       = (const float*)d_in[0];
  const float* w_conv0 = (const float*)d_in[1];
  const float* w_conv1 = (const float*)d_in[2];
  const float* w_conv2 = (const float*)d_in[3];
  const float* w_conv3 = (const float*)d_in[4];
  const float* w_mp    = (const float*)d_in[5];
  const float* qkv_w   = (const float*)d_in[6];
  const float* qkv_b   = (const float*)d_in[7];
  const float* se_w1   = (const float*)d_in[8];
  const float* se_w2   = (const float*)d_in[9];
  const float* w_bott  = (const float*)d_in[10];
  const float* bn_g    = (const float*)d_in[11];
  const float* bn_b    = (const float*)d_in[12];

  float* ws    = (float*)d_ws;
  float* F     = ws + OFF_F;       // also reused as z after xc is built
  float* gaps  = ws + OFF_GAPS;
  float* att   = ws + OFF_ATT;
  float* y     = ws + OFF_Y;
  float* stats = ws + OFF_STATS;
  float* out   = (float*)d_out;    // doubles as xc scratch between k4 and k6

  // 1) branch feature maps via f32 WMMA GEMMs
  conv5_kernel<<<dim3(LLEN / 64, B_SZ, NFT), 128, 0, stream>>>(
      x, w_conv0, w_conv1, w_conv2, w_conv3, w_mp, F);

  // 2) per-branch GAP
  gap_kernel<<<dim3(NF, B_SZ, NFT), 256, 0, stream>>>(F, gaps);

  // 3) attention + SE gates
  attse_kernel<<<B_SZ, 128, 0, stream>>>(gaps, qkv_w, qkv_b, se_w1, se_w2,
                                         att, y);

  // 4) xc = SE-scaled attention mix of branches (into d_out as scratch)
  xcbuild_kernel<<<(B_SZ * NF * LLEN) / 256, 256, 0, stream>>>(F, att, y, out);

  // 5) 1x1 bottleneck GEMM via f32 WMMA (z overwrites the dead F region)
  bott_kernel<<<dim3(LLEN / 64, B_SZ, NC / 128), 128, 0, stream>>>(
      out, w_bott, F);

  // 5b) BN batch statistics
  bnstats_kernel<<<NC, 256, 0, stream>>>(F, stats);

  // 6) BN + ReLU -> final output
  bnfinal_kernel<<<(B_SZ * NC * LLEN) / 256, 256, 0, stream>>>(
      F, stats, bn_g, bn_b, out);
}